// Motion_Channel_Excitation_41936060678820
// MI455X (gfx1250) — compile-verified
//
#include <hip/hip_runtime.h>
#include <hip/hip_bf16.h>
#include <math.h>
#include <stdint.h>

typedef __attribute__((ext_vector_type(2))) float v2f;
typedef __attribute__((ext_vector_type(4))) float v4f;
typedef __attribute__((ext_vector_type(8))) float v8f;

#define N_  32
#define C_  256
#define T_  300
#define V_  25
#define R_  16
#define NCT (N_*C_*T_)   /* 2,457,600 rows of length V_ */
#define NRT (N_*R_*T_)   /* 153,600   */
#define TILES_PER_N ((T_ + 15) / 16)   /* 19 */
#define ROWS_PER_BLK 256               /* rows per block in K1/K4 */
#define ELEMS_PER_BLK (ROWS_PER_BLK * V_)   /* 6400 floats = 25.6 KB */

// ---------------------------------------------------------------------------
// K1: xbar[row] = mean over v of x[row, 0..24]
// Block stages 6400 contiguous floats into LDS via async b128 global->LDS
// copies (ASYNCcnt), then each thread reduces its own row from LDS.
// NCT is divisible by 256, and ELEMS_PER_BLK divisible by 4, so no tails.
// ---------------------------------------------------------------------------
__global__ void k_vmean(const float* __restrict__ x, float* __restrict__ xbar) {
    __shared__ float buf[ELEMS_PER_BLK];
    const int tid = threadIdx.x;
    const size_t base = (size_t)blockIdx.x * ELEMS_PER_BLK;

    const unsigned long long gbase = (unsigned long long)(const void*)(x + base);
    const unsigned ldsbase = (unsigned)(uintptr_t)buf;

    // 1600 x b128 async copies, fully coalesced, no VGPR round-trip
    for (int i = tid; i < ELEMS_PER_BLK / 4; i += 256) {
        unsigned voff    = (unsigned)i * 16u;          // global byte offset
        unsigned ldsaddr = ldsbase + (unsigned)i * 16u; // LDS byte address
        asm volatile("global_load_async_to_lds_b128 %0, %1, %2"
                     :: "v"(ldsaddr), "v"(voff), "s"(gbase) : "memory");
    }
    asm volatile("s_wait_asynccnt 0x0" ::: "memory");
    __syncthreads();

    float s = 0.f;
    #pragma unroll
    for (int v = 0; v < V_; ++v) s += buf[tid * V_ + v];
    xbar[(size_t)blockIdx.x * ROWS_PER_BLK + tid] = s * (1.0f / V_);
}

// ---------------------------------------------------------------------------
// K2: ybar[n,r,t] = BN( sq_w(16x256) @ xbar[n,:,t] + sq_b )
// One wave per 16x16 output tile, 64 x V_WMMA_F32_16X16X4_F32 over c=256.
// ---------------------------------------------------------------------------
__global__ void k_squeeze_wmma(const float* __restrict__ xbar,
                               const float* __restrict__ sq_w,
                               const float* __restrict__ sq_b,
                               const float* __restrict__ bn_gamma,
                               const float* __restrict__ bn_beta,
                               const float* __restrict__ bn_mean,
                               const float* __restrict__ bn_var,
                               float* __restrict__ ybar) {
    const int lane  = threadIdx.x & 31;
    const int wave  = threadIdx.x >> 5;
    const int gw    = blockIdx.x * (blockDim.x >> 5) + wave;
    if (gw >= N_ * TILES_PER_N) return;

    const int n    = gw / TILES_PER_N;
    const int col0 = (gw % TILES_PER_N) * 16;

    const int m      = lane & 15;              // row (r) for A, col (t) for B/D
    const int khalf  = (lane >> 4) << 1;       // 0 or 2
    int col = col0 + m; if (col > T_ - 1) col = T_ - 1;   // clamp tail loads

    const float* xb = xbar + (size_t)n * (C_ * T_);

    v8f acc = {0.f, 0.f, 0.f, 0.f, 0.f, 0.f, 0.f, 0.f};
    for (int k0 = 0; k0 < C_; k0 += 4) {
        const int kb = k0 + khalf;
        v2f a, b;
        a.x = sq_w[m * C_ + kb];
        a.y = sq_w[m * C_ + kb + 1];
        b.x = xb[(size_t)kb       * T_ + col];
        b.y = xb[(size_t)(kb + 1) * T_ + col];
        acc = __builtin_amdgcn_wmma_f32_16x16x4_f32(
                  false, a, false, b, (short)0, acc, false, false);
    }

    const bool valid = (col0 + m) < T_;
    const int  rbase = (lane >> 4) << 3;
    #pragma unroll
    for (int j = 0; j < 8; ++j) {
        const int r = j + rbase;
        const float inv_std = bn_gamma[r] * rsqrtf(bn_var[r] + 1e-5f);
        const float val = (acc[j] + sq_b[r] - bn_mean[r]) * inv_std + bn_beta[r];
        if (valid)
            ybar[(size_t)n * (R_ * T_) + r * T_ + (col0 + m)] = val;
    }
}

// ---------------------------------------------------------------------------
// K3: score[n,c,t] = sigmoid( exp_w(256x16) @ m[n,:,t] + exp_b )
// m[r'] = conv1_b[r'] + conv1_w @ ybar[:,t+1] - ybar[r',t]  (t%3 != 2), else 0.
// Phase 1: build 16x16 M tile in LDS. Phase 2: expand GEMM via WMMA.
// ---------------------------------------------------------------------------
__global__ void k_score_wmma(const float* __restrict__ ybar,
                             const float* __restrict__ conv1_w,
                             const float* __restrict__ conv1_b,
                             const float* __restrict__ exp_w,
                             const float* __restrict__ exp_b,
                             float* __restrict__ score) {
    __shared__ float Mlds[16 * 16];
    const int j0 = blockIdx.x * 16;          // 600 blocks * 16 = 9600 = N_*T_

    {
        const int tid  = threadIdx.x;
        const int r    = tid >> 4;
        const int colL = tid & 15;
        const int jg   = j0 + colL;
        const int n    = jg / T_;
        const int t    = jg - n * T_;
        float mval = 0.f;
        if ((t % 3) != 2) {
            const float* yb = ybar + (size_t)n * (R_ * T_);
            float a = conv1_b[r] - yb[r * T_ + t];
            #pragma unroll
            for (int r2 = 0; r2 < 16; ++r2)
                a += conv1_w[r * 16 + r2] * yb[r2 * T_ + (t + 1)];
            mval = a;
        }
        Mlds[r * 16 + colL] = mval;
    }
    __syncthreads();

    const int lane  = threadIdx.x & 31;
    const int wave  = threadIdx.x >> 5;
    const int mrow  = lane & 15;
    const int khalf = (lane >> 4) << 1;
    const int jg    = j0 + mrow;
    const int n     = jg / T_;
    const int t     = jg - n * T_;
    const int rbase = (lane >> 4) << 3;

    #pragma unroll
    for (int sub = 0; sub < 2; ++sub) {
        const int c0 = (wave * 2 + sub) * 16;
        v8f acc = {0.f, 0.f, 0.f, 0.f, 0.f, 0.f, 0.f, 0.f};
        #pragma unroll
        for (int k0 = 0; k0 < 16; k0 += 4) {
            const int kb = k0 + khalf;
            v2f a, b;
            a.x = exp_w[(c0 + mrow) * 16 + kb];
            a.y = exp_w[(c0 + mrow) * 16 + kb + 1];
            b.x = Mlds[kb       * 16 + mrow];
            b.y = Mlds[(kb + 1) * 16 + mrow];
            acc = __builtin_amdgcn_wmma_f32_16x16x4_f32(
                      false, a, false, b, (short)0, acc, false, false);
        }
        #pragma unroll
        for (int jj = 0; jj < 8; ++jj) {
            const int c = c0 + jj + rbase;
            float sv = acc[jj] + exp_b[c];
            sv = 1.0f / (1.0f + __expf(-sv));
            score[(size_t)n * (C_ * T_) + c * T_ + t] = sv;
        }
    }
}

// ---------------------------------------------------------------------------
// K4: out[row, v] = x[row, v] * score[row]
// Block handles 256 rows = 6400 elements. Scores staged in LDS; x read and
// out written with fully coalesced float4 accesses; per-element row index via
// cheap u32 div-by-25 (strength-reduced to mul/shift).
// ---------------------------------------------------------------------------
__global__ void k_apply(const float* __restrict__ x,
                        const float* __restrict__ score,
                        float* __restrict__ out) {
    __shared__ float s_lds[ROWS_PER_BLK];
    const int tid = threadIdx.x;
    const size_t rowbase = (size_t)blockIdx.x * ROWS_PER_BLK;
    s_lds[tid] = score[rowbase + tid];
    __syncthreads();

    const size_t base = rowbase * V_;
    const v4f* xp = (const v4f*)(x + base);
    v4f*       op = (v4f*)(out + base);

    for (int i = tid; i < ELEMS_PER_BLK / 4; i += 256) {
        v4f xv = xp[i];
        const unsigned e = (unsigned)i * 4u;
        v4f ov;
        ov.x = xv.x * s_lds[(e + 0u) / 25u];
        ov.y = xv.y * s_lds[(e + 1u) / 25u];
        ov.z = xv.z * s_lds[(e + 2u) / 25u];
        ov.w = xv.w * s_lds[(e + 3u) / 25u];
        op[i] = ov;
    }
}

// ---------------------------------------------------------------------------
extern "C" void kernel_launch(void* const* d_in, const int* in_sizes, int n_in,
                              void* d_out, int out_size, void* d_ws, size_t ws_size,
                              hipStream_t stream) {
    (void)in_sizes; (void)n_in; (void)out_size; (void)ws_size;

    const float* x        = (const float*)d_in[0];
    const float* sq_w     = (const float*)d_in[1];
    const float* sq_b     = (const float*)d_in[2];
    const float* bn_gamma = (const float*)d_in[3];
    const float* bn_beta  = (const float*)d_in[4];
    const float* bn_mean  = (const float*)d_in[5];
    const float* bn_var   = (const float*)d_in[6];
    const float* conv1_w  = (const float*)d_in[7];
    const float* conv1_b  = (const float*)d_in[8];
    const float* exp_w    = (const float*)d_in[9];
    const float* exp_b    = (const float*)d_in[10];
    float*       out      = (float*)d_out;

    float* xbar  = (float*)d_ws;            // NCT floats
    float* ybar  = xbar + NCT;              // NRT floats
    float* score = ybar + NRT;              // NCT floats

    // K1: v-mean, 2,457,600 rows / 256 rows per block = 9600 blocks
    k_vmean<<<NCT / ROWS_PER_BLK, 256, 0, stream>>>(x, xbar);

    // K2: squeeze+BN GEMM, 32*19 = 608 wave-tiles, 8 waves/block
    k_squeeze_wmma<<<(N_ * TILES_PER_N + 7) / 8, 256, 0, stream>>>(
        xbar, sq_w, sq_b, bn_gamma, bn_beta, bn_mean, bn_var, ybar);

    // K3: conv1-diff-mean + expand GEMM + sigmoid, 9600/16 = 600 blocks
    k_score_wmma<<<(N_ * T_) / 16, 256, 0, stream>>>(
        ybar, conv1_w, conv1_b, exp_w, exp_b, score);

    // K4: broadcast multiply, coalesced float4
    k_apply<<<NCT / ROWS_PER_BLK, 256, 0, stream>>>(x, score, out);
}